// u_i_GCN_21028159881433
// MI455X (gfx1250) — compile-verified
//
#include <hip/hip_runtime.h>

// ---------------------------------------------------------------------------
// LightGCN propagation on MI455X (gfx1250).
//   out = (x0 + A@x0 + A@(A@x0) + A@(A@(A@x0))) / 4,  A = weighted scatter-add
// L2-resident problem (h = 12.8 MB, edges = 19.2 MB vs 192 MB L2).
// One wave32 per edge-slot: lane = feature column (d == 32 == wave width).
// Edge tiles staged into LDS with gfx1250 async global->LDS loads.
// ---------------------------------------------------------------------------

namespace {
constexpr int kD     = 32;    // feature dim == wave32 width
constexpr int kBlock = 256;   // 8 wave32 per workgroup
constexpr int kTile  = 256;   // edges staged per block
constexpr int kWaves = kBlock / 32;
}

// gfx1250 async global->LDS builtin takes typed AS-qualified pointers:
//   param0: __device__ int*  (address_space(1))
//   param1: __shared__ int*  (address_space(3))
typedef __attribute__((address_space(1))) int gint_t;  // global int
typedef __attribute__((address_space(3))) int lint_t;  // LDS int

#if __has_builtin(__builtin_amdgcn_global_load_async_to_lds_b32)
#define GCN_HAS_ASYNC_LDS 1
#else
#define GCN_HAS_ASYNC_LDS 0
#endif

__device__ __forceinline__ void gcn_wait_async0() {
#if __has_builtin(__builtin_amdgcn_s_wait_asynccnt)
  __builtin_amdgcn_s_wait_asynccnt(0);
#else
  asm volatile("s_wait_asynccnt 0" ::: "memory");
#endif
}

// acc = x0 ; h = 0
__global__ __launch_bounds__(kBlock) void gcn_init(const float* __restrict__ x0,
                                                   float* __restrict__ acc,
                                                   float* __restrict__ h, int n) {
  int i = blockIdx.x * kBlock + threadIdx.x;
  if (i < n) {
    acc[i] = x0[i];
    h[i] = 0.0f;
  }
}

// acc += h ; z = 0   (fused so each 12.8 MB array is touched once)
__global__ __launch_bounds__(kBlock) void gcn_add_zero(float* __restrict__ acc,
                                                       const float* __restrict__ h,
                                                       float* __restrict__ z, int n) {
  int i = blockIdx.x * kBlock + threadIdx.x;
  if (i < n) {
    acc[i] += h[i];
    z[i] = 0.0f;
  }
}

// acc = (acc + h) * 1/(LAYERS+1)
__global__ __launch_bounds__(kBlock) void gcn_finalize(float* __restrict__ acc,
                                                       const float* __restrict__ h, int n) {
  int i = blockIdx.x * kBlock + threadIdx.x;
  if (i < n) acc[i] = (acc[i] + h[i]) * 0.25f;
}

// Scatter SpMM layer: hout[dst,:] += w * hin[src,:]
// Edge tile (src,dst,w) staged into LDS via async global->LDS copies; then
// each wave walks its edges: coalesced 128B row gather + 32 f32 atomics.
__global__ __launch_bounds__(kBlock) void gcn_spmm_scatter(
    const int* __restrict__ esrc, const int* __restrict__ edst,
    const float* __restrict__ ew, const float* __restrict__ hin,
    float* __restrict__ hout, int E) {
  __shared__ int   sSrc[kTile];
  __shared__ int   sDst[kTile];
  __shared__ float sW[kTile];

  const int  t    = threadIdx.x;
  const long base = (long)blockIdx.x * kTile;
  const long rem  = (long)E - base;
  const int  cnt  = rem < (long)kTile ? (int)rem : kTile;

  if (cnt == kTile) {
#if GCN_HAS_ASYNC_LDS
    // Each wave issues 3 async b32 loads covering 32 contiguous edges each.
    __builtin_amdgcn_global_load_async_to_lds_b32(
        (gint_t*)(void*)(esrc + base + t), (lint_t*)&sSrc[t], 0, 0);
    __builtin_amdgcn_global_load_async_to_lds_b32(
        (gint_t*)(void*)(edst + base + t), (lint_t*)&sDst[t], 0, 0);
    __builtin_amdgcn_global_load_async_to_lds_b32(
        (gint_t*)(void*)(ew + base + t), (lint_t*)(void*)&sW[t], 0, 0);
    gcn_wait_async0();  // this wave's ASYNCcnt drained; barrier covers the rest
#else
    sSrc[t] = esrc[base + t];
    sDst[t] = edst[base + t];
    sW[t]   = ew[base + t];
#endif
  } else {
    // Ragged tail (unreachable for E = 1.6M, kept for safety).
    if (t < cnt) {
      sSrc[t] = esrc[base + t];
      sDst[t] = edst[base + t];
      sW[t]   = ew[base + t];
    } else {
      sSrc[t] = 0;
      sDst[t] = 0;
      sW[t]   = 0.0f;
    }
  }
  __syncthreads();

  const int wave = t >> 5;
  const int lane = t & 31;
  for (int i = wave; i < cnt; i += kWaves) {
    // Prefetch the next edge's source row (128B) one iteration ahead.
    const int ni = i + kWaves;
    if (ni < cnt) {
      __builtin_prefetch(&hin[(size_t)sSrc[ni] * kD + lane], 0, 3);
    }
    const int   s = sSrc[i];
    const int   d = sDst[i];
    const float w = sW[i];
    const float v = w * hin[(size_t)s * kD + lane];
    atomicAdd(&hout[(size_t)d * kD + lane], v);  // global_atomic_add_f32, no-return
  }
}

extern "C" void kernel_launch(void* const* d_in, const int* in_sizes, int n_in,
                              void* d_out, int out_size, void* d_ws, size_t ws_size,
                              hipStream_t stream) {
  const float* x0  = (const float*)d_in[0];  // [N, 32]
  const int*   es  = (const int*)d_in[1];    // [E]
  const int*   ed  = (const int*)d_in[2];    // [E]
  const float* ewt = (const float*)d_in[3];  // [E]

  const int ND = in_sizes[0];  // N * 32
  const int E  = in_sizes[1];

  float* acc = (float*)d_out;
  float* hA  = (float*)d_ws;
  float* hB  = hA + (size_t)ND;  // ws usage: 2 * 12.8 MB

  const dim3 blk(kBlock);
  const dim3 egrid((unsigned)((ND + kBlock - 1) / kBlock));
  const dim3 sgrid((unsigned)((E + kTile - 1) / kTile));

  // acc = x0 ; hA = 0
  gcn_init<<<egrid, blk, 0, stream>>>(x0, acc, hA, ND);
  // layer 1: hA = A @ x0
  gcn_spmm_scatter<<<sgrid, blk, 0, stream>>>(es, ed, ewt, x0, hA, E);
  // acc += hA ; hB = 0
  gcn_add_zero<<<egrid, blk, 0, stream>>>(acc, hA, hB, ND);
  // layer 2: hB = A @ hA
  gcn_spmm_scatter<<<sgrid, blk, 0, stream>>>(es, ed, ewt, hA, hB, E);
  // acc += hB ; hA = 0
  gcn_add_zero<<<egrid, blk, 0, stream>>>(acc, hB, hA, ND);
  // layer 3: hA = A @ hB
  gcn_spmm_scatter<<<sgrid, blk, 0, stream>>>(es, ed, ewt, hB, hA, E);
  // acc = (acc + hA) / 4
  gcn_finalize<<<egrid, blk, 0, stream>>>(acc, hA, ND);
}